// Hbv_2h_87514253623609
// MI455X (gfx1250) — compile-verified
//
#include <hip/hip_runtime.h>
#include <math.h>

#define T_STEPS 1440
#define N_GRID  8000
#define N_OUT   32
#define LENF    72
#define HBV_DT  (1.0f / 24.0f)
#define NEARZ   1e-5f

typedef __attribute__((ext_vector_type(2))) float v2f;
typedef __attribute__((ext_vector_type(8))) float v8f;

// ---------------------------------------------------------------------------
// Phase 1: HBV recurrence. One thread per grid cell, sequential over time.
// Memory-bound streaming of x_phy (138 MB); all params descaled into regs.
// ---------------------------------------------------------------------------
__global__ void hbv_kernel(const float* __restrict__ x_phy,
                           const float* __restrict__ ac_all,
                           const float* __restrict__ p1,
                           float* __restrict__ Q) {
  const int n = blockIdx.x * blockDim.x + threadIdx.x;
  if (n >= N_GRID) return;
  const float* pr = p1 + (size_t)n * 19;
  const float BETA   = 1.0f   + pr[0]  * 5.0f;
  const float FC     = 50.0f  + pr[1]  * 950.0f;
  const float K0     = 0.05f  + pr[2]  * 0.85f;
  const float K1     = 0.01f  + pr[3]  * 0.49f;
  const float K2     = 0.001f + pr[4]  * 0.199f;
  const float LP     = 0.2f   + pr[5]  * 0.8f;
  const float PERC   =          pr[6]  * 10.0f;
  const float UZL    =          pr[7]  * 100.0f;
  const float TT     = -2.5f  + pr[8]  * 5.0f;
  const float CFMAX  = 0.5f   + pr[9]  * 9.5f;
  const float CFR    =          pr[10] * 0.1f;
  const float CWH    =          pr[11] * 0.2f;
  const float BETAET = 0.3f   + pr[12] * 4.7f;
  const float Cpar   =          pr[13];
  const float AC     =          pr[15] * 2500.0f;
  const float F0     = 120.0f + pr[16] * 2760.0f;
  const float FMIN   =          pr[17];
  const float ALPHA  = 0.5f   + pr[18] * 4.5f;

  const float ac = ac_all[n];
  const float fmax_ = HBV_DT * F0 *
      (FMIN + (1.0f - FMIN) * __expf(-__powf(ac / (AC + 1.0f), 1.0f / ALPHA)));

  float SP = 1e-3f, MW = 1e-3f, SM = 1e-3f, SUZ = 1e-3f, SLZ = 1e-3f;
  const float* xp = x_phy + (size_t)n * 3;

  for (int t = 0; t < T_STEPS; ++t) {
    const size_t base = (size_t)t * (size_t)N_GRID * 3;
    // speculative stream prefetch (emits global_prefetch_b8 on gfx1250)
    __builtin_prefetch(&xp[base + (size_t)8 * N_GRID * 3], 0, 0);
    const float P   = xp[base + 0];
    const float Tm  = xp[base + 1];
    const float PET = xp[base + 2];

    const float rain = (Tm >= TT) ? P : 0.0f;
    const float snow = P - rain;
    SP += snow;
    const float melt = fminf(fmaxf(CFMAX * HBV_DT * (Tm - TT), 0.0f), SP);
    MW += melt; SP -= melt;
    const float refr = fminf(fmaxf(CFR * CFMAX * HBV_DT * (TT - Tm), 0.0f), MW);
    SP += refr; MW -= refr;
    const float tosoil = fmaxf(MW - CWH * SP, 0.0f);
    MW -= tosoil;
    const float win   = rain + tosoil;
    const float infil = fminf(win, fmax_);
    const float qie   = win - infil;
    float sw = __powf(SM / FC, BETA);
    sw = fminf(fmaxf(sw, 0.0f), 1.0f);
    const float rech = infil * sw;
    SM += infil - rech;
    const float excess = fmaxf(SM - FC, 0.0f);
    SM -= excess;
    const float cap = fminf(SLZ, Cpar * SLZ * (1.0f - fminf(SM / FC, 1.0f)));
    SM += cap; SLZ -= cap;
    float ef = __powf(SM / (LP * FC), BETAET);
    ef = fminf(fmaxf(ef, 0.0f), 1.0f);
    const float et = fminf(PET * ef, SM);
    SM = fmaxf(SM - et, NEARZ);
    SUZ += rech + excess;
    const float perc = fminf(SUZ, PERC * HBV_DT);
    SUZ -= perc;
    const float q0 = K0 * HBV_DT * fmaxf(SUZ - UZL, 0.0f);
    SUZ -= q0;
    const float q1 = K1 * HBV_DT * SUZ;
    SUZ -= q1;
    SLZ += perc;
    const float q2 = K2 * HBV_DT * SLZ;
    SLZ -= q2;

    Q[(size_t)t * N_GRID + n] = qie + q0 + q1 + q2;  // coalesced over n
  }
}

// ---------------------------------------------------------------------------
// Phase 2: routing weights, stored k-major W[k][n] for coalesced conv loads.
// Two-pass (sum then normalize) to keep VGPR count low.
// ---------------------------------------------------------------------------
__global__ void weights_kernel(const float* __restrict__ p1,
                               const float* __restrict__ p2,
                               float* __restrict__ Wk) {
  const int n = blockIdx.x * blockDim.x + threadIdx.x;
  if (n >= N_GRID) return;
  const float RT   = p1[(size_t)n * 19 + 14] * 20.0f;
  const float ra   = p2[(size_t)n * 3 + 0] * 5.0f;
  const float rb   = p2[(size_t)n * 3 + 1] * 12.0f;
  const float rtau = p2[(size_t)n * 3 + 2] * 48.0f;
  const float lag = rtau + RT;
  const float a = fmaxf(ra, 0.01f);
  const float b = fmaxf(rb, 0.01f);

  float sum = 0.0f;
  for (int k = 0; k < LENF; ++k) {
    const float s  = (k + 0.5f) - lag;
    const float sm = fmaxf(s, 1e-6f);
    const float lw = (a - 1.0f) * __logf(sm) - sm / b;
    sum += (s > 0.0f) ? __expf(lw) : 0.0f;
  }
  const float inv = 1.0f / (sum + 1e-8f);
  for (int k = 0; k < LENF; ++k) {
    const float s  = (k + 0.5f) - lag;
    const float sm = fmaxf(s, 1e-6f);
    const float lw = (a - 1.0f) * __logf(sm) - sm / b;
    const float wv = (s > 0.0f) ? __expf(lw) : 0.0f;
    Wk[(size_t)k * N_GRID + n] = wv * inv;
  }
}

// ---------------------------------------------------------------------------
// Phase 3a: transpose outlet_topo (32 x 8000) -> topoT (8000 x 32)
// so WMMA B-fragment loads are contiguous.
// ---------------------------------------------------------------------------
__global__ void transpose_topo_kernel(const float* __restrict__ topo,
                                      float* __restrict__ topoT) {
  const int idx = blockIdx.x * blockDim.x + threadIdx.x;
  if (idx >= N_GRID * N_OUT) return;
  const int n = idx >> 5;
  const int o = idx & 31;
  topoT[idx] = topo[(size_t)o * N_GRID + n];
}

// ---------------------------------------------------------------------------
// Phase 3b: per-cell causal conv (length-72 per-channel filter), LDS-tiled.
// 64 t-rows x 64 cells per block; also folds in areas[n].
// ---------------------------------------------------------------------------
__global__ void conv_route_kernel(const float* __restrict__ Q,
                                  const float* __restrict__ Wk,
                                  const float* __restrict__ areas,
                                  float* __restrict__ Rr) {
  __shared__ float Qs[64 + LENF - 1][64];  // 135 x 64
  __shared__ float Wl[LENF][64];
  const int n0  = blockIdx.x * 64;
  const int t0  = blockIdx.y * 64;
  const int tid = threadIdx.x;

  for (int idx = tid; idx < (64 + LENF - 1) * 64; idx += 256) {
    const int r = idx >> 6, c = idx & 63;
    const int gt = t0 - (LENF - 1) + r;
    Qs[r][c] = (gt >= 0 && gt < T_STEPS) ? Q[(size_t)gt * N_GRID + n0 + c] : 0.0f;
  }
  for (int idx = tid; idx < LENF * 64; idx += 256) {
    const int k = idx >> 6, c = idx & 63;
    Wl[k][c] = Wk[(size_t)k * N_GRID + n0 + c];
  }
  __syncthreads();

  const int c  = tid & 63;
  const int rg = tid >> 6;  // 0..3, thread owns rows rg, rg+4, ..., rg+60
  float acc[16];
#pragma unroll
  for (int i = 0; i < 16; ++i) acc[i] = 0.0f;

  for (int k = 0; k < LENF; ++k) {
    const float wv = Wl[k][c];
#pragma unroll
    for (int i = 0; i < 16; ++i)
      acc[i] += Qs[rg + 4 * i + (LENF - 1) - k][c] * wv;
  }

  const float area = areas[n0 + c];
#pragma unroll
  for (int i = 0; i < 16; ++i) {
    const int t = t0 + rg + 4 * i;
    if (t < T_STEPS) Rr[(size_t)t * N_GRID + n0 + c] = acc[i] * area;
  }
}

// ---------------------------------------------------------------------------
// Phase 4: out (1440x32) = Rr (1440x8000) x topoT (8000x32) in fp32 WMMA.
// One wave per 16x16 output tile; A-tiles (16 t-rows x 64 k) staged in LDS
// with coalesced float4 loads; K consumed in steps of 4 by
// V_WMMA_F32_16X16X4_F32.
// ---------------------------------------------------------------------------
__global__ void gemm_out_kernel(const float* __restrict__ Rr,
                                const float* __restrict__ topoT,
                                float* __restrict__ out) {
  __shared__ __align__(16) float As[16][64];
  const int o0   = blockIdx.x * 16;
  const int t0   = blockIdx.y * 16;
  const int lane = threadIdx.x;          // 0..31 (wave32)
  const int m    = lane & 15;            // row (A) / col (B) within tile
  const int kgrp = (lane >> 4) * 2;      // lanes 0-15 -> K{0,1}; 16-31 -> K{2,3}

  v8f acc = {};

  for (int kk = 0; kk < N_GRID; kk += 64) {
    // stage 16x64 A tile: 256 float4s, 8 per lane, contiguous per row
#pragma unroll
    for (int i = 0; i < 8; ++i) {
      const int idx = i * 32 + lane;     // 0..255
      const int r   = idx >> 4;          // 0..15
      const int seg = idx & 15;          // 0..15  (16B chunks)
      const float4 q = *reinterpret_cast<const float4*>(
          &Rr[(size_t)(t0 + r) * N_GRID + kk + seg * 4]);
      *reinterpret_cast<float4*>(&As[r][seg * 4]) = q;
    }
    __syncthreads();
#pragma unroll
    for (int kb = 0; kb < 64; kb += 4) {
      v2f a, b;
      a.x = As[m][kb + kgrp];
      a.y = As[m][kb + kgrp + 1];
      const float* bp = &topoT[(size_t)(kk + kb + kgrp) * N_OUT + o0 + m];
      b.x = bp[0];
      b.y = bp[N_OUT];
      acc = __builtin_amdgcn_wmma_f32_16x16x4_f32(
          /*neg_a=*/false, a, /*neg_b=*/false, b,
          /*c_mod=*/(short)0, acc, /*reuse_a=*/false, /*reuse_b=*/false);
    }
    __syncthreads();
  }

  // D layout: VGPR i -> M=i (lanes 0-15) / M=i+8 (lanes 16-31), N = lane&15
#pragma unroll
  for (int i = 0; i < 8; ++i) {
    const int mm = (lane < 16) ? i : (i + 8);
    out[(size_t)(t0 + mm) * N_OUT + o0 + (lane & 15)] = acc[i];
  }
}

// ---------------------------------------------------------------------------
extern "C" void kernel_launch(void* const* d_in, const int* in_sizes, int n_in,
                              void* d_out, int out_size, void* d_ws, size_t ws_size,
                              hipStream_t stream) {
  const float* x_phy  = (const float*)d_in[0];  // (1440, 8000, 3)
  const float* ac_all = (const float*)d_in[1];  // (8000,)
  // d_in[2] = elev_all (unused by reference math)
  const float* topo   = (const float*)d_in[3];  // (32, 8000)
  const float* areas  = (const float*)d_in[4];  // (8000,)
  const float* p1     = (const float*)d_in[5];  // (8000, 19)
  const float* p2     = (const float*)d_in[6];  // (8000, 3)
  float* out = (float*)d_out;                   // (1440, 32)

  float* Q     = (float*)d_ws;                        // [T][N]
  float* Rr    = Q  + (size_t)T_STEPS * N_GRID;       // [T][N]
  float* Wk    = Rr + (size_t)T_STEPS * N_GRID;       // [72][N]
  float* topoT = Wk + (size_t)LENF * N_GRID;          // [N][32]

  hbv_kernel<<<(N_GRID + 255) / 256, 256, 0, stream>>>(x_phy, ac_all, p1, Q);
  weights_kernel<<<(N_GRID + 255) / 256, 256, 0, stream>>>(p1, p2, Wk);
  transpose_topo_kernel<<<(N_GRID * N_OUT + 255) / 256, 256, 0, stream>>>(topo, topoT);
  conv_route_kernel<<<dim3(N_GRID / 64, (T_STEPS + 63) / 64), 256, 0, stream>>>(
      Q, Wk, areas, Rr);
  gemm_out_kernel<<<dim3(N_OUT / 16, T_STEPS / 16), 32, 0, stream>>>(Rr, topoT, out);
}